// DotProdAttn_87677462381134
// MI455X (gfx1250) — compile-verified
//
#include <hip/hip_runtime.h>

typedef __attribute__((ext_vector_type(2))) float v2f;
typedef __attribute__((ext_vector_type(8))) float v8f;

#define NB 32
#define NL 8192
#define ND 512

// ---------------------------------------------------------------------------
// Kernel 1: scores[b,l] = sum_d q[b,d] * k[b,l,d]  via V_WMMA_F32_16X16X4_F32
// Block: 256 threads = 8 waves; each wave owns 16 rows of K.
// Grid: (NL/128, NB).
// A = q replicated across M rows (A layout per ISA: lanes 0-15 -> k=0,1;
// lanes 16-31 -> k=2,3). B = 16x4 chunk of K^T staged through LDS so the
// global loads stay fully coalesced (b128). Column n of the 16x16 accumulator
// is the score of K row n; every M row is identical, read M=0 (acc[0]).
// ---------------------------------------------------------------------------
__global__ __launch_bounds__(256) void scores_wmma_kernel(
    const float* __restrict__ q, const float* __restrict__ k,
    float* __restrict__ scores) {
  __shared__ float qs[ND];            // 2 KB
  __shared__ float ks[8][16 * 68];    // 8 waves * 4352 B, +4 float row pad
  const int tid  = threadIdx.x;
  const int wave = tid >> 5;
  const int lane = tid & 31;
  const int b    = blockIdx.y;
  const int row0 = blockIdx.x * 128 + wave * 16;

  for (int i = tid; i < ND; i += 256) qs[i] = q[b * ND + i];
  __syncthreads();

  const float* kb = k + (size_t)b * NL * ND;
  const int h2 = (lane >> 4) * 2;  // 0 for lanes 0-15, 2 for lanes 16-31
  const int m  = lane & 15;
  v8f acc = {};

  for (int d0 = 0; d0 < ND; d0 += 64) {
    // Stage 16 rows x 64 cols of K, coalesced 512B-per-instruction loads.
#pragma unroll
    for (int i = 0; i < 8; ++i) {
      const int t = i * 128 + lane * 4;
      const int r = t >> 6;
      const int c = t & 63;
      const float4 vv =
          *(const float4*)(kb + (size_t)(row0 + r) * ND + d0 + c);
      float* dst = &ks[wave][r * 68 + c];
      dst[0] = vv.x; dst[1] = vv.y; dst[2] = vv.z; dst[3] = vv.w;
    }
    __syncthreads();
#pragma unroll
    for (int kk = 0; kk < 16; ++kk) {
      const int k4 = kk * 4;
      v2f afrag, bfrag;
      afrag.x = qs[d0 + k4 + h2];
      afrag.y = qs[d0 + k4 + h2 + 1];
      bfrag.x = ks[wave][m * 68 + k4 + h2];
      bfrag.y = ks[wave][m * 68 + k4 + h2 + 1];
      acc = __builtin_amdgcn_wmma_f32_16x16x4_f32(
          false, afrag, false, bfrag, (short)0, acc, false, false);
    }
    __syncthreads();
  }
  if (lane < 16) scores[(size_t)b * NL + row0 + lane] = acc[0];
}

// ---------------------------------------------------------------------------
// Reductions (wave32).
// ---------------------------------------------------------------------------
__device__ __forceinline__ float waveMax(float v) {
#pragma unroll
  for (int o = 16; o > 0; o >>= 1) v = fmaxf(v, __shfl_xor(v, o, 32));
  return v;
}
__device__ __forceinline__ float waveSum(float v) {
#pragma unroll
  for (int o = 16; o > 0; o >>= 1) v += __shfl_xor(v, o, 32);
  return v;
}
__device__ __forceinline__ float blockMax(float v, float* red, int wave,
                                          int lane) {
  v = waveMax(v);
  if (lane == 0) red[wave] = v;
  __syncthreads();
  v = waveMax(red[lane]);  // 32 waves -> red[0..31]; all waves redundantly
  __syncthreads();
  return v;
}
__device__ __forceinline__ float blockSum(float v, float* red, int wave,
                                          int lane) {
  v = waveSum(v);
  if (lane == 0) red[wave] = v;
  __syncthreads();
  v = waveSum(red[lane]);
  __syncthreads();
  return v;
}

// ---------------------------------------------------------------------------
// Kernel 2: double softmax over L per batch. One 1024-thread block per batch,
// whole row resident in 32 KB LDS. Second softmax's max is exactly 1/sum1
// (max of exp(x - max) == 1), so only three reductions total.
// ---------------------------------------------------------------------------
__global__ __launch_bounds__(1024) void softmax2_kernel(
    const float* __restrict__ scores, float* __restrict__ att_out) {
  __shared__ float s[NL];    // 32 KB
  __shared__ float red[32];
  const int b = blockIdx.x;
  const int tid = threadIdx.x;
  const int wave = tid >> 5, lane = tid & 31;
  const float* sc = scores + (size_t)b * NL;

  float lmax = -3.402823466e38f;
  for (int i = tid; i < NL; i += 1024) {
    const float x = sc[i];
    s[i] = x;
    lmax = fmaxf(lmax, x);
  }
  __syncthreads();
  const float bmax = blockMax(lmax, red, wave, lane);

  float lsum = 0.f;
  for (int i = tid; i < NL; i += 1024) {
    const float e = __expf(s[i] - bmax);
    s[i] = e;
    lsum += e;
  }
  __syncthreads();
  const float inv1 = 1.f / blockSum(lsum, red, wave, lane);

  // second softmax: w = e*inv1, max(w) = inv1 exactly
  float lsum2 = 0.f;
  for (int i = tid; i < NL; i += 1024) {
    const float f = __expf(s[i] * inv1 - inv1);
    s[i] = f;
    lsum2 += f;
  }
  __syncthreads();
  const float inv2 = 1.f / blockSum(lsum2, red, wave, lane);

  float* dst = att_out + (size_t)b * NL;
  for (int i = tid; i < NL; i += 1024) dst[i] = s[i] * inv2;
}

// ---------------------------------------------------------------------------
// Kernel 3: ctx[b,d] = sum_l att[b,l] * V[b,l,d]. One block per batch,
// 512 threads = 4 L-strips x 128 float4 columns; V streamed once, coalesced
// 2KB per row. att row cached in LDS.
// ---------------------------------------------------------------------------
__global__ __launch_bounds__(512) void ctx_kernel(
    const float* __restrict__ att, const float* __restrict__ v,
    float* __restrict__ ctx) {
  __shared__ float a[NL];             // 32 KB
  __shared__ float part[4][ND];       // 8 KB strip partials
  const int b = blockIdx.x;
  const int tid = threadIdx.x;
  for (int i = tid; i < NL; i += 512) a[i] = att[(size_t)b * NL + i];
  __syncthreads();

  const int strip = tid >> 7;   // 0..3
  const int c4    = tid & 127;  // float4 column group
  const float* vb = v + (size_t)b * NL * ND;
  float4 acc = {0.f, 0.f, 0.f, 0.f};
  for (int l = strip; l < NL; l += 4) {
    const float4 vv = *(const float4*)(vb + (size_t)l * ND + c4 * 4);
    const float w = a[l];
    acc.x += w * vv.x; acc.y += w * vv.y;
    acc.z += w * vv.z; acc.w += w * vv.w;
  }
  float* p = &part[strip][c4 * 4];
  p[0] = acc.x; p[1] = acc.y; p[2] = acc.z; p[3] = acc.w;
  __syncthreads();
  if (strip == 0) {
#pragma unroll
    for (int s2 = 1; s2 < 4; ++s2) {
      const float* pp = &part[s2][c4 * 4];
      acc.x += pp[0]; acc.y += pp[1]; acc.z += pp[2]; acc.w += pp[3];
    }
    float* o = ctx + (size_t)b * ND + c4 * 4;
    o[0] = acc.x; o[1] = acc.y; o[2] = acc.z; o[3] = acc.w;
  }
}

// ---------------------------------------------------------------------------
extern "C" void kernel_launch(void* const* d_in, const int* in_sizes, int n_in,
                              void* d_out, int out_size, void* d_ws,
                              size_t ws_size, hipStream_t stream) {
  const float* q = (const float*)d_in[0];   // [B, D]
  const float* k = (const float*)d_in[1];   // [B, L, D]
  const float* v = (const float*)d_in[2];   // [B, L, D]
  float* out = (float*)d_out;
  float* ctx = out;                 // [B, D] first in tuple
  float* att = out + NB * ND;       // [B, 1, L] second in tuple
  float* scores = (float*)d_ws;     // B*L floats = 1 MB scratch

  scores_wmma_kernel<<<dim3(NL / 128, NB), 256, 0, stream>>>(q, k, scores);
  softmax2_kernel<<<NB, 1024, 0, stream>>>(scores, att);
  ctx_kernel<<<NB, 512, 0, stream>>>(att, v, ctx);
}